// NequIPModel_22076131902171
// MI455X (gfx1250) — compile-verified
//
#include <hip/hip_runtime.h>
#include <cmath>

// ---------------------------------------------------------------------------
// NequIP energy+forces on gfx1250 (MI455X). All GEMM-shaped contractions use
// V_WMMA_F32_16X16X4_F32 (fp32 WMMA) with LDS-tiled fragments.
// Round 3: kill branch-sinking around WMMA operand loads:
//  (a) weights staged into zero-padded LDS tiles -> unguarded LDS fragments;
//  (b) Y/g fragments use index-select into a guaranteed-zero LDS slot ->
//      unconditional loads with a v_cndmask on the address only.
// ---------------------------------------------------------------------------

typedef float v2f __attribute__((ext_vector_type(2)));
typedef float v8f __attribute__((ext_vector_type(8)));

#define NB      16
#define NN      512
#define NBATCH  2
#define NODES   (NBATCH*NN)
#define FSTR    320
#define NF      ((size_t)NODES*FSTR)
#define RC      5.0f
#define INVS    0.18257418583505536f   /* 1/sqrt(30) */
#define PI_F    3.14159265358979323846f
#define RBC     0.6324555320336759f    /* sqrt(2/Rc) */

static __device__ const int C_[4]   = {64,32,16, 8};
static __device__ const int M_[4]   = { 1, 3, 5, 7};
static __device__ const int OFF_[4] = { 0,64,160,240};
static __device__ const int PK_[4]  = { 0,64, 96,112};
static __device__ const int YO_[4]  = { 0, 1,  4,  9};
static __device__ const int NCT_[4] = { 4, 2,  1,  1};
static __device__ const int D_[4]   = {64,96, 80, 56};

struct BlockW {
  const float* WrA[4]; const float* Ws[4]; const float* WrB[4];
  const float* V[4];   const float* LW[4]; const float* LB[4];
};

// ---------------- WMMA helpers (fp32 16x16x4) ------------------------------
static __device__ inline v8f vzero8(){ v8f v; for(int i=0;i<8;i++) v[i]=0.f; return v; }

static __device__ inline v8f wmma4(v2f a, v2f b, v8f c){
  return __builtin_amdgcn_wmma_f32_16x16x4_f32(false, a, false, b, (short)0, c, false, false);
}
static __device__ inline int mini(int a, int b){ return a<b?a:b; }

// A fragment: element (row r, k) at p[r*rs + k*ks].
static __device__ inline v2f ldA(const float* __restrict__ p, int rs, int ks, int k0){
  int ln = threadIdx.x & 31; int r = ln & 15; int kb = k0 + ((ln >> 4) << 1);
  v2f a; a.x = p[r*rs + kb*ks]; a.y = p[r*rs + (kb+1)*ks]; return a;
}
// B fragment: element (k, col) at p[k*ks + col]   (unguarded; padded tiles)
static __device__ inline v2f ldB(const float* __restrict__ p, int ks, int k0){
  int ln = threadIdx.x & 31; int c = ln & 15; int kb = k0 + ((ln >> 4) << 1);
  v2f b; b.x = p[kb*ks + c]; b.y = p[(kb+1)*ks + c]; return b;
}
// Transposed B fragment: element (k, col) at p[col*cs + k] (unguarded; padded)
static __device__ inline v2f ldBTu(const float* __restrict__ p, int cs, int k0){
  int ln = threadIdx.x & 31; int c = ln & 15; int kb = k0 + ((ln >> 4) << 1);
  v2f b; b.x = p[c*cs + kb]; b.y = p[c*cs + kb+1]; return b;
}
// B fragment from Y tile [16][17] (col 16 == 0): B(k=j, col=m)=Yt[j][YO+m].
static __device__ inline v2f ldB_Y(const float (*Yt)[17], int YO, int M, int k0){
  int ln = threadIdx.x & 31; int m = ln & 15; int kb = k0 + ((ln >> 4) << 1);
  int mi = (m<M) ? (YO+m) : 16;                   // zero-slot select (address)
  v2f b; b.x = Yt[kb][mi]; b.y = Yt[kb+1][mi]; return b;
}
// A fragment from Y tile: A(row=j, k=m) = Yt[j][YO+m].
static __device__ inline v2f ldA_Y(const float (*Yt)[17], int YO, int M, int k0){
  int ln = threadIdx.x & 31; int j = ln & 15; int kb = k0 + ((ln >> 4) << 1);
  int i1 = (kb  <M) ? (YO+kb)   : 16;
  int i2 = (kb+1<M) ? (YO+kb+1) : 16;
  v2f a; a.x = Yt[j][i1]; a.y = Yt[j][i2]; return a;
}
// zero-slot index into g (LDS, size FSTR+1, g[FSTR]==0)
static __device__ inline int gidx(int OFF, int c, int m, int C, int M){
  return (c<C && m<M) ? (OFF + c*M + m) : FSTR;
}
// B fragment from g: B(k=c, col=m) = g[OFF+(c0+k)*M+m].
static __device__ inline v2f ldB_g(const float* __restrict__ g, int OFF, int c0, int C, int M, int k0){
  int ln = threadIdx.x & 31; int m = ln & 15; int kb = k0 + ((ln >> 4) << 1);
  v2f b; b.x = g[gidx(OFF,c0+kb,m,C,M)]; b.y = g[gidx(OFF,c0+kb+1,m,C,M)]; return b;
}
// B fragment from g: B(k=m, col=c) = g[OFF+(c0+c)*M+m].
static __device__ inline v2f ldB_gT(const float* __restrict__ g, int OFF, int c0, int C, int M, int k0){
  int ln = threadIdx.x & 31; int c = ln & 15; int kb = k0 + ((ln >> 4) << 1);
  v2f b; b.x = g[gidx(OFF,c0+c,kb,C,M)]; b.y = g[gidx(OFF,c0+c,kb+1,C,M)]; return b;
}

// Stage rows x cols matrix into LDS tile with stride ds, zero-padded to
// ceil16(rows) x ceil16(cols). Done once per WG per matrix (branch OK here).
static __device__ inline void stage_mat(float* dst, const float* __restrict__ src,
                                        int rows, int cols, int ds){
  int R16 = (rows+15)&~15, C16 = (cols+15)&~15;
  for(int idx=threadIdx.x; idx<R16*C16; idx+=blockDim.x){
    int r = idx / C16, c = idx % C16;
    dst[r*ds+c] = (r<rows && c<cols) ? src[r*cols+c] : 0.f;
  }
}

// ---------------- geometry / basis -----------------------------------------
static __device__ inline float ycomp(int m, float x, float y, float z){
  switch(m){
    case 0: return 1.f;
    case 1: return x; case 2: return y; case 3: return z;
    case 4: return x*y; case 5: return y*z; case 6: return 3.f*z*z-1.f;
    case 7: return x*z; case 8: return x*x-y*y;
    case 9: return y*(3.f*x*x-y*y); case 10: return x*y*z;
    case 11: return y*(5.f*z*z-1.f); case 12: return z*(5.f*z*z-3.f);
    case 13: return x*(5.f*z*z-1.f); case 14: return z*(x*x-y*y);
    default: return x*(x*x-3.f*y*y);
  }
}
static __device__ inline void ygrad(int m, float x, float y, float z, float g[3]){
  g[0]=0.f; g[1]=0.f; g[2]=0.f;
  switch(m){
    case 0: break;
    case 1: g[0]=1.f; break; case 2: g[1]=1.f; break; case 3: g[2]=1.f; break;
    case 4: g[0]=y; g[1]=x; break;
    case 5: g[1]=z; g[2]=y; break;
    case 6: g[2]=6.f*z; break;
    case 7: g[0]=z; g[2]=x; break;
    case 8: g[0]=2.f*x; g[1]=-2.f*y; break;
    case 9: g[0]=6.f*x*y; g[1]=3.f*x*x-3.f*y*y; break;
    case 10:g[0]=y*z; g[1]=x*z; g[2]=x*y; break;
    case 11:g[1]=5.f*z*z-1.f; g[2]=10.f*y*z; break;
    case 12:g[2]=15.f*z*z-3.f; break;
    case 13:g[0]=5.f*z*z-1.f; g[2]=10.f*x*z; break;
    case 14:g[0]=2.f*x*z; g[1]=-2.f*y*z; g[2]=x*x-y*y; break;
    default:g[0]=3.f*x*x-3.f*y*y; g[1]=-6.f*x*y; break;
  }
}
// Stage R (16 basis) and Y (16 sph comps + zero col) for pairs (i, j0..j0+15).
static __device__ inline void stage_pair(const float* __restrict__ pos, int b, int i, int j0,
                                         float (*Rt)[16], float (*Yt)[17],
                                         float* pd, float (*pu)[3], int* pval){
  int t = threadIdx.x; int j = t & 15; int grp = t >> 4;  // 0..7
  int jj = j0 + j;
  float px = pos[((size_t)(b*NN+i))*3+0];
  float py = pos[((size_t)(b*NN+i))*3+1];
  float pz = pos[((size_t)(b*NN+i))*3+2];
  float qx = pos[((size_t)(b*NN+jj))*3+0];
  float qy = pos[((size_t)(b*NN+jj))*3+1];
  float qz = pos[((size_t)(b*NN+jj))*3+2];
  float rx = qx-px, ry = qy-py, rz = qz-pz;      // rij = pos_j - pos_i
  float d  = __builtin_sqrtf(rx*rx+ry*ry+rz*rz+1e-12f);
  float dn = fmaxf(d, 1e-6f);
  float ux = rx/dn, uy = ry/dn, uz = rz/dn;
  int valid = (d < RC) && (i != jj);
  float env = 0.5f*(__cosf(PI_F*fminf(d,RC)/RC)+1.f);
  float pre = valid ? RBC*env/dn : 0.f;
  for(int s=0;s<2;s++){
    int k = grp + 8*s;
    Rt[j][k] = pre*__sinf((float)(k+1)*PI_F*d/RC);
    Yt[j][k] = ycomp(k, ux, uy, uz);
  }
  if(grp==0){
    Yt[j][16]=0.f;                                // zero slot
    pd[j]=d; pu[j][0]=ux; pu[j][1]=uy; pu[j][2]=uz; pval[j]=valid;
  }
}
// Pack radial weights into LDS: W[k*128 + PK_l + c], zero-padded.
static __device__ inline void stage_W(const BlockW& bw, float* WA, float* WB){
  for(int idx=threadIdx.x; idx<16*128; idx+=blockDim.x){ WA[idx]=0.f; WB[idx]=0.f; }
  __syncthreads();
  for(int l=0;l<4;l++){
    int C = C_[l], tot = 16*C;
    for(int idx=threadIdx.x; idx<tot; idx+=blockDim.x){
      int k = idx / C, c = idx % C;
      WA[k*128 + PK_[l] + c] = bw.WrA[l][idx];
      WB[k*128 + PK_[l] + c] = bw.WrB[l][idx];
    }
  }
}
static __device__ inline float siluf(float x){ return x/(1.f+__expf(-x)); }
static __device__ inline float dsiluf(float x){
  float s = 1.f/(1.f+__expf(-x)); return s*(1.f + x*(1.f-s));
}

// ---------------- utility kernels ------------------------------------------
__global__ void k_zero(float* __restrict__ p, size_t n){
  size_t i = (size_t)blockIdx.x*blockDim.x + threadIdx.x;
  size_t st = (size_t)gridDim.x*blockDim.x;
  for(; i<n; i+=st) p[i]=0.f;
}
__global__ void k_embed(const int* __restrict__ z, const float* __restrict__ emb,
                        float* __restrict__ f0){
  int node = blockIdx.x*blockDim.x + threadIdx.x;
  if(node >= NODES) return;
  int zi = z[node];
  for(int c=0;c<FSTR;c++) f0[(size_t)node*FSTR+c] = (c<64) ? emb[zi*64+c] : 0.f;
}
__global__ void k_init_g(const float* __restrict__ ow, float* __restrict__ g){
  int node = blockIdx.x*blockDim.x + threadIdx.x;
  if(node >= NODES) return;
  for(int c=0;c<FSTR;c++) g[(size_t)node*FSTR+c] = (c<64) ? ow[c] : 0.f;
}
__global__ void k_energy(const float* __restrict__ f, const float* __restrict__ ow,
                         const float* __restrict__ ob, float* __restrict__ E){
  int node = blockIdx.x*blockDim.x + threadIdx.x;
  if(node >= NODES) return;
  float e = ob[0];
  for(int c=0;c<64;c++) e += f[(size_t)node*FSTR+c]*ow[c];
  atomicAdd(&E[node/NN], e);
}

// ---------------- sA = s0 @ W_s ; fB = feats @ V  (WMMA) --------------------
__global__ __launch_bounds__(32) void k_sA_fB(const float* __restrict__ feats, BlockW bw,
                                              float* __restrict__ sA, float* __restrict__ fB){
  __shared__ float Wl[64*64];
  int n0 = blockIdx.x*16; int ln = threadIdx.x & 31;
  for(int l=0;l<4;l++){
    int C=C_[l], M=M_[l], OFF=OFF_[l], PK=PK_[l];
    int nct = (C+15)/16;
    stage_mat(Wl, bw.Ws[l], 64, C, 64);           // [64][C] zero-padded cols
    __syncthreads();
    for(int ct=0; ct<nct; ct++){
      int c0 = ct*16; int ncol = (C-c0<16)?(C-c0):16;
      v8f acc = vzero8();
      for(int kk=0;kk<16;kk++){
        v2f a  = ldA(feats + (size_t)n0*FSTR, FSTR, 1, kk*4);
        v2f bb = ldB(&Wl[c0], 64, kk*4);
        acc = wmma4(a, bb, acc);
      }
      for(int r=0;r<8;r++){
        int row = r + ((ln>>4)<<3); int c = ln & 15;
        if(c < ncol) sA[(size_t)(n0+row)*128 + PK + c0 + c] = acc[r];
      }
    }
    __syncthreads();
    stage_mat(Wl, bw.V[l], C, C, 64);             // [C][C] zero-padded
    __syncthreads();
    for(int m=0;m<M;m++){
      for(int ct=0; ct<nct; ct++){
        int c0 = ct*16; int ncol = (C-c0<16)?(C-c0):16;
        v8f acc = vzero8();
        for(int kk=0;kk<C/4;kk++){
          v2f a  = ldA(feats + (size_t)n0*FSTR + OFF + m, FSTR, M, kk*4);
          v2f bb = ldB(&Wl[c0], 64, kk*4);
          acc = wmma4(a, bb, acc);
        }
        for(int r=0;r<8;r++){
          int row = r + ((ln>>4)<<3); int c = ln & 15;
          if(c < ncol) fB[(size_t)(n0+row)*FSTR + OFF + (c0+c)*M + m] = acc[r];
        }
      }
    }
    __syncthreads();
  }
}

// ---------------- fused pair convolution (forward) --------------------------
__global__ __launch_bounds__(128) void k_conv(const float* __restrict__ pos,
                                              const float* __restrict__ sA,
                                              const float* __restrict__ fB,
                                              BlockW bw, float* __restrict__ convo){
  __shared__ float Rt[16][16], Yt[16][17];
  __shared__ float sAt[16][128];
  __shared__ float fBt[16][FSTR];
  __shared__ float tbuf[4][16][16];
  __shared__ float WA[2048], WB[2048];
  __shared__ float pd[16], pu[16][3]; __shared__ int pval[16];
  int node = blockIdx.x; int b = node/NN; int i = node%NN;
  int wid = threadIdx.x>>5; int ln = threadIdx.x&31; int l = wid;
  stage_W(bw, WA, WB);
  __syncthreads();
  int C=C_[l], M=M_[l], OFF=OFF_[l], PK=PK_[l], YO=YO_[l], nct=NCT_[l];
  v8f acc[4]; for(int q=0;q<4;q++) acc[q]=vzero8();
  for(int jt=0;jt<NN/16;jt++){
    int j0 = jt*16;
    stage_pair(pos, b, i, j0, Rt, Yt, pd, pu, pval);
    for(int idx=threadIdx.x; idx<2048; idx+=128)
      (&sAt[0][0])[idx] = sA[(size_t)(b*NN+j0)*128 + idx];
    for(int idx=threadIdx.x; idx<16*FSTR; idx+=128)
      (&fBt[0][0])[idx] = fB[(size_t)(b*NN+j0)*FSTR + idx];
    __syncthreads();
    for(int ct=0;ct<nct;ct++){
      int c0 = ct*16;
      // wA tile = R @ W_radA  (WMMA, K=16)
      v8f wa = vzero8();
      for(int kk=0;kk<4;kk++)
        wa = wmma4(ldA(&Rt[0][0],16,1,kk*4), ldB(&WA[PK+c0],128,kk*4), wa);
      // t = wA .* sA  -> LDS (reindex), then acc += t^T @ Y_l  (WMMA)
      for(int r=0;r<8;r++){
        int j=r+((ln>>4)<<3); int c=ln&15;
        tbuf[wid][j][c] = wa[r]*sAt[j][PK+c0+c];
      }
      for(int kk=0;kk<4;kk++){
        v2f a  = ldA(&tbuf[wid][0][0], 1, 16, kk*4);      // A(row=c,k=j)
        v2f bb = ldB_Y(Yt, YO, M, kk*4);                  // B(k=j,col=m), zero-slot
        acc[ct] = wmma4(a, bb, acc[ct]);
      }
      // wB tile = R @ W_radB, then path B: acc[c,m] += sum_j wB[j,c]*fB[j,c,m]
      v8f wb = vzero8();
      for(int kk=0;kk<4;kk++)
        wb = wmma4(ldA(&Rt[0][0],16,1,kk*4), ldB(&WB[PK+c0],128,kk*4), wb);
      for(int r=0;r<8;r++){
        int j=r+((ln>>4)<<3); int c=ln&15;
        tbuf[wid][j][c] = wb[r];
      }
      for(int r=0;r<8;r++){
        int cl = r + ((ln>>4)<<3); int cab = c0 + cl; int m = ln & 15;
        if(cab<C && m<M){
          float s=0.f;
          for(int j=0;j<16;j++) s += tbuf[wid][j][cl]*fBt[j][OFF + cab*M + m];
          acc[ct][r] += s;
        }
      }
    }
    __syncthreads();
  }
  for(int ct=0;ct<nct;ct++)
    for(int r=0;r<8;r++){
      int cab = ct*16 + r + ((ln>>4)<<3); int m = ln & 15;
      if(cab<C && m<M) convo[(size_t)node*FSTR + OFF + cab*M + m] = acc[ct][r]*INVS;
    }
}

// ---------------- lin + gated residual (forward) ----------------------------
__global__ __launch_bounds__(32) void k_lin(const float* __restrict__ convo,
                                            const float* __restrict__ fin,
                                            BlockW bw, float* __restrict__ hpre,
                                            float* __restrict__ zpre,
                                            float* __restrict__ fout){
  __shared__ float Wl[96*96];
  int n0 = blockIdx.x*16; int ln = threadIdx.x & 31;
  for(int l=0;l<4;l++){
    int D=D_[l], OFF=OFF_[l];
    stage_mat(Wl, bw.LW[l], D, D, 96);
    __syncthreads();
    for(int ct=0; ct<(D+15)/16; ct++){
      int c0 = ct*16; int ncol = (D-c0<16)?(D-c0):16;
      v8f acc = vzero8();
      for(int kk=0;kk<D/4;kk++){
        v2f a  = ldA(convo + (size_t)n0*FSTR + OFF, FSTR, 1, kk*4);
        v2f bb = ldB(&Wl[c0], 96, kk*4);
        acc = wmma4(a, bb, acc);
      }
      for(int r=0;r<8;r++){
        int row = r + ((ln>>4)<<3); int c = ln & 15;
        if(c < ncol){
          size_t gi = (size_t)(n0+row)*FSTR + OFF + c0 + c;
          float h = acc[r] + bw.LB[l][c0+c];
          hpre[gi] = h;
          float zv = fin[gi] + siluf(h);
          zpre[gi] = zv;
          fout[gi] = siluf(zv);
        }
      }
    }
    __syncthreads();
  }
}

// ---------------- backward: lin layer ---------------------------------------
__global__ __launch_bounds__(32) void k_lin_bwd(const float* __restrict__ gout,
                                                const float* __restrict__ zpre,
                                                const float* __restrict__ hpre, BlockW bw,
                                                float* __restrict__ gconv,
                                                float* __restrict__ gin){
  __shared__ float Wl[96*96];
  __shared__ float dht[16][96];
  int n0 = blockIdx.x*16; int ln = threadIdx.x & 31;
  for(int l=0;l<4;l++){
    int D=D_[l], OFF=OFF_[l];
    for(int idx=ln; idx<16*D; idx+=32){
      int row = idx/D, c = idx%D;
      size_t gi = (size_t)(n0+row)*FSTR + OFF + c;
      float dz = gout[gi]*dsiluf(zpre[gi]);
      gin[gi]  = dz;                       // residual path
      dht[row][c] = dz*dsiluf(hpre[gi]);
    }
    stage_mat(Wl, bw.LW[l], D, D, 96);     // padded: transposed reads safe
    __syncthreads();
    for(int ct=0; ct<(D+15)/16; ct++){
      int c0 = ct*16; int ncol = (D-c0<16)?(D-c0):16;
      v8f acc = vzero8();
      for(int kk=0;kk<D/4;kk++){
        v2f a  = ldA(&dht[0][0], 96, 1, kk*4);
        v2f bb = ldBTu(&Wl[c0*96], 96, kk*4);          // lin_w^T, unguarded
        acc = wmma4(a, bb, acc);
      }
      for(int r=0;r<8;r++){
        int row = r + ((ln>>4)<<3); int c = ln & 15;
        if(c < ncol) gconv[(size_t)(n0+row)*FSTR + OFF + c0 + c] = acc[r];
      }
    }
    __syncthreads();
  }
}

// ---------------- backward: forces (per-pair dR/dY -> analytic chain) -------
__global__ __launch_bounds__(128) void k_conv_bwdA(const float* __restrict__ pos,
                                                   const float* __restrict__ sA,
                                                   const float* __restrict__ fB,
                                                   const float* __restrict__ gconv,
                                                   BlockW bw, float* __restrict__ F){
  __shared__ float Rt[16][16], Yt[16][17];
  __shared__ float sAt[16][128];
  __shared__ float fBt[16][FSTR];
  __shared__ float tbuf[4][16][16];
  __shared__ float WA[2048], WB[2048];
  __shared__ float gI[FSTR+1];
  __shared__ float dRs[4][16][16], dYs[4][16][16];
  __shared__ float pd[16], pu[16][3]; __shared__ int pval[16];
  __shared__ float fred[16][3];
  int node = blockIdx.x; int b = node/NN; int i = node%NN;
  int wid = threadIdx.x>>5; int ln = threadIdx.x&31; int l = wid;
  stage_W(bw, WA, WB);
  for(int idx=threadIdx.x; idx<FSTR; idx+=128) gI[idx] = gconv[(size_t)node*FSTR+idx]*INVS;
  if(threadIdx.x==0) gI[FSTR]=0.f;               // zero slot
  __syncthreads();
  int C=C_[l], M=M_[l], OFF=OFF_[l], PK=PK_[l], YO=YO_[l], nct=NCT_[l];
  float facc0=0.f, facc1=0.f, facc2=0.f;
  for(int jt=0;jt<NN/16;jt++){
    int j0 = jt*16;
    stage_pair(pos, b, i, j0, Rt, Yt, pd, pu, pval);
    for(int idx=threadIdx.x; idx<2048; idx+=128)
      (&sAt[0][0])[idx] = sA[(size_t)(b*NN+j0)*128 + idx];
    for(int idx=threadIdx.x; idx<16*FSTR; idx+=128)
      (&fBt[0][0])[idx] = fB[(size_t)(b*NN+j0)*FSTR + idx];
    __syncthreads();
    v8f dracc = vzero8(), dyacc = vzero8();
    for(int ct=0;ct<nct;ct++){
      int c0 = ct*16;
      // wA tile
      v8f wa = vzero8();
      for(int kk=0;kk<4;kk++)
        wa = wmma4(ldA(&Rt[0][0],16,1,kk*4), ldB(&WA[PK+c0],128,kk*4), wa);
      // tA = wA.*sA -> dY += tA @ g   (dY[j,m] = sum_c tA[j,c] g[c,m])
      for(int r=0;r<8;r++){
        int j=r+((ln>>4)<<3); int c=ln&15;
        tbuf[wid][j][c] = wa[r]*sAt[j][PK+c0+c];
      }
      for(int kk=0;kk<4;kk++){
        v2f a  = ldA(&tbuf[wid][0][0], 16, 1, kk*4);          // A(row=j,k=c)
        v2f bb = ldB_g(gI, OFF, c0, C, M, kk*4);              // B(k=c,col=m)
        dyacc = wmma4(a, bb, dyacc);
      }
      // q[j,c] = sum_m Y[j,m] g[c,m]  (WMMA, zero-slot guards)
      v8f q = vzero8();
      for(int kk=0;kk<4;kk++){
        v2f a  = ldA_Y(Yt, YO, M, kk*4);                      // A(row=j,k=m)
        v2f bb = ldB_gT(gI, OFF, c0, C, M, kk*4);             // B(k=m,col=c)
        q = wmma4(a, bb, q);
      }
      // dwA = q.*sA -> dR += dwA @ W_radA^T
      for(int r=0;r<8;r++){
        int j=r+((ln>>4)<<3); int c=ln&15;
        tbuf[wid][j][c] = q[r]*sAt[j][PK+c0+c];
      }
      for(int kk=0;kk<4;kk++){
        v2f a  = ldA(&tbuf[wid][0][0], 16, 1, kk*4);
        v2f bb;
        { int kbas=ln&15; int kb=kk*4+((ln>>4)<<1);
          bb.x = WA[kbas*128+PK+c0+kb];
          bb.y = WA[kbas*128+PK+c0+kb+1]; }
        dracc = wmma4(a, bb, dracc);
      }
      // dwB[j,c] = sum_m g[c,m] fB[j,c,m]  (zero-slot makes invalid c give 0)
      for(int r=0;r<8;r++){
        int j=r+((ln>>4)<<3); int c=ln&15; int cab=c0+c;
        int cu = mini(cab, C-1);
        float s=0.f;
        for(int m=0;m<M;m++) s += gI[gidx(OFF,cab,m,C,M)]*fBt[j][OFF+cu*M+m];
        tbuf[wid][j][c] = s;
      }
      for(int kk=0;kk<4;kk++){
        v2f a  = ldA(&tbuf[wid][0][0], 16, 1, kk*4);
        v2f bb;
        { int kbas=ln&15; int kb=kk*4+((ln>>4)<<1);
          bb.x = WB[kbas*128+PK+c0+kb];
          bb.y = WB[kbas*128+PK+c0+kb+1]; }
        dracc = wmma4(a, bb, dracc);
      }
    }
    for(int r=0;r<8;r++){
      int j=r+((ln>>4)<<3); int c=ln&15;
      dRs[wid][j][c]=dracc[r]; dYs[wid][j][c]=dyacc[r];
    }
    __syncthreads();
    for(int idx=threadIdx.x; idx<256; idx+=128){
      int j=idx>>4, k=idx&15;
      dRs[0][j][k] += dRs[1][j][k]+dRs[2][j][k]+dRs[3][j][k];
      dYs[0][j][k] += dYs[1][j][k]+dYs[2][j][k]+dYs[3][j][k];
    }
    __syncthreads();
    if(threadIdx.x < 16){
      int j = threadIdx.x;
      if(pval[j]){
        float d=pd[j]; float ux=pu[j][0],uy=pu[j][1],uz=pu[j][2];
        float inv_d = 1.f/fmaxf(d,1e-6f);
        float env  = 0.5f*(__cosf(PI_F*d/RC)+1.f);
        float denv = -0.5f*(PI_F/RC)*__sinf(PI_F*d/RC);
        float gd = 0.f;
        for(int k=0;k<16;k++){
          float ak = (float)(k+1)*PI_F/RC;
          float sn=__sinf(ak*d), cs=__cosf(ak*d);
          float rb  = RBC*sn*inv_d;
          float drb = RBC*(ak*cs*inv_d - sn*inv_d*inv_d);
          gd += dRs[0][j][k]*(drb*env + rb*denv);
        }
        float gu0=0.f,gu1=0.f,gu2=0.f;
        for(int m=0;m<16;m++){
          float g3[3]; ygrad(m,ux,uy,uz,g3);
          float w = dYs[0][j][m];
          gu0 += w*g3[0]; gu1 += w*g3[1]; gu2 += w*g3[2];
        }
        float gdu = gu0*ux+gu1*uy+gu2*uz;
        float dr0 = gd*ux + (gu0-gdu*ux)*inv_d;
        float dr1 = gd*uy + (gu1-gdu*uy)*inv_d;
        float dr2 = gd*uz + (gu2-gdu*uz)*inv_d;
        facc0 += dr0; facc1 += dr1; facc2 += dr2;     // F_i += dE/dr
        int jj = j0+j;
        atomicAdd(&F[(size_t)(b*NN+jj)*3+0], -dr0);   // F_j -= dE/dr
        atomicAdd(&F[(size_t)(b*NN+jj)*3+1], -dr1);
        atomicAdd(&F[(size_t)(b*NN+jj)*3+2], -dr2);
      }
    }
    __syncthreads();
  }
  if(threadIdx.x < 16){ fred[threadIdx.x][0]=facc0; fred[threadIdx.x][1]=facc1; fred[threadIdx.x][2]=facc2; }
  __syncthreads();
  if(threadIdx.x == 0){
    float s0=0.f,s1=0.f,s2=0.f;
    for(int j=0;j<16;j++){ s0+=fred[j][0]; s1+=fred[j][1]; s2+=fred[j][2]; }
    atomicAdd(&F[(size_t)(b*NN+i)*3+0], s0);
    atomicAdd(&F[(size_t)(b*NN+i)*3+1], s1);
    atomicAdd(&F[(size_t)(b*NN+i)*3+2], s2);
  }
}

// ---------------- backward: dsA / dfB (atomics-free, per j-tile owner) ------
__global__ __launch_bounds__(128) void k_conv_bwdB(const float* __restrict__ pos,
                                                   const float* __restrict__ gconv,
                                                   BlockW bw, float* __restrict__ dsA,
                                                   float* __restrict__ dfB){
  __shared__ float Rt[16][16], Yt[16][17];
  __shared__ float tbuf[4][16][16];
  __shared__ float WA[2048], WB[2048];
  __shared__ float gI[FSTR+1];
  __shared__ float dfBa[16][FSTR];
  __shared__ float pd[16], pu[16][3]; __shared__ int pval[16];
  int blk = blockIdx.x; int b = blk/(NN/16); int jt = blk%(NN/16); int j0 = jt*16;
  int wid = threadIdx.x>>5; int ln = threadIdx.x&31; int l = wid;
  stage_W(bw, WA, WB);
  for(int idx=threadIdx.x; idx<16*FSTR; idx+=128) (&dfBa[0][0])[idx] = 0.f;
  if(threadIdx.x==0) gI[FSTR]=0.f;               // zero slot (persists)
  __syncthreads();
  int C=C_[l], M=M_[l], OFF=OFF_[l], PK=PK_[l], YO=YO_[l], nct=NCT_[l];
  v8f dsacc[4]; for(int q=0;q<4;q++) dsacc[q]=vzero8();
  for(int i=0;i<NN;i++){
    stage_pair(pos, b, i, j0, Rt, Yt, pd, pu, pval);
    for(int idx=threadIdx.x; idx<FSTR; idx+=128) gI[idx] = gconv[(size_t)(b*NN+i)*FSTR+idx]*INVS;
    __syncthreads();
    for(int ct=0;ct<nct;ct++){
      int c0 = ct*16;
      v8f wa = vzero8();
      for(int kk=0;kk<4;kk++)
        wa = wmma4(ldA(&Rt[0][0],16,1,kk*4), ldB(&WA[PK+c0],128,kk*4), wa);
      v8f q = vzero8();
      for(int kk=0;kk<4;kk++){
        v2f a  = ldA_Y(Yt, YO, M, kk*4);
        v2f bb = ldB_gT(gI, OFF, c0, C, M, kk*4);
        q = wmma4(a, bb, q);
      }
      for(int r=0;r<8;r++) dsacc[ct][r] += wa[r]*q[r];       // dsA += wA .* (Y g^T)
      v8f wb = vzero8();
      for(int kk=0;kk<4;kk++)
        wb = wmma4(ldA(&Rt[0][0],16,1,kk*4), ldB(&WB[PK+c0],128,kk*4), wb);
      for(int r=0;r<8;r++){
        int j=r+((ln>>4)<<3); int c=ln&15;
        tbuf[wid][j][c] = wb[r];
      }
      for(int r=0;r<8;r++){
        int cl = r + ((ln>>4)<<3); int cab = c0 + cl; int m = ln & 15;
        if(cab<C && m<M){
          float gv = gI[OFF + cab*M + m];
          for(int j=0;j<16;j++) dfBa[j][OFF + cab*M + m] += tbuf[wid][j][cl]*gv;
        }
      }
    }
    __syncthreads();
  }
  for(int ct=0;ct<nct;ct++)
    for(int r=0;r<8;r++){
      int j=r+((ln>>4)<<3); int c=ln&15; int cab=ct*16+c;
      if(cab<C) dsA[(size_t)(b*NN+j0+j)*128 + PK + cab] = dsacc[ct][r];
    }
  __syncthreads();
  for(int idx=threadIdx.x; idx<16*FSTR; idx+=128)
    dfB[(size_t)(b*NN+j0)*FSTR + idx] = (&dfBa[0][0])[idx];
}

// ---------------- backward: dsA/dfB -> input feature grads (WMMA) -----------
__global__ __launch_bounds__(32) void k_sAfB_bwd(const float* __restrict__ dsA,
                                                 const float* __restrict__ dfB,
                                                 BlockW bw, float* __restrict__ gin){
  __shared__ float Wl[64*64];
  int n0 = blockIdx.x*16; int ln = threadIdx.x & 31;
  // ds0 = sum_l dsA_l @ W_s_l^T  -> add into gin l0 channels
  v8f acc4[4]; for(int q=0;q<4;q++) acc4[q]=vzero8();
  for(int l=0;l<4;l++){
    int C=C_[l], PK=PK_[l];
    stage_mat(Wl, bw.Ws[l], 64, C, 64);
    __syncthreads();
    for(int ct=0;ct<4;ct++){
      int c0 = ct*16;
      for(int kk=0;kk<C/4;kk++){
        v2f a  = ldA(dsA + (size_t)n0*128 + PK, 128, 1, kk*4);
        v2f bb = ldBTu(&Wl[c0*64], 64, kk*4);   // (k=cl, col=c64)=Ws[c64][cl]
        acc4[ct] = wmma4(a, bb, acc4[ct]);
      }
    }
    __syncthreads();
  }
  for(int ct=0;ct<4;ct++)
    for(int r=0;r<8;r++){
      int row = r + ((ln>>4)<<3); int c = ln & 15;
      gin[(size_t)(n0+row)*FSTR + ct*16 + c] += acc4[ct][r];
    }
  // dfeats_l = dfB_l @ V_l^T (per m) -> add into gin
  for(int l=0;l<4;l++){
    int C=C_[l], M=M_[l], OFF=OFF_[l];
    stage_mat(Wl, bw.V[l], C, C, 64);
    __syncthreads();
    for(int m=0;m<M;m++)
      for(int ct=0; ct<(C+15)/16; ct++){
        int c0 = ct*16; int ncol = (C-c0<16)?(C-c0):16;
        v8f acc = vzero8();
        for(int kk=0;kk<C/4;kk++){
          v2f a  = ldA(dfB + (size_t)n0*FSTR + OFF + m, FSTR, M, kk*4);
          v2f bb = ldBTu(&Wl[c0*64], 64, kk*4);  // (k=d, col=c)=V[c][d]
          acc = wmma4(a, bb, acc);
        }
        for(int r=0;r<8;r++){
          int row = r + ((ln>>4)<<3); int c = ln & 15;
          if(c < ncol) gin[(size_t)(n0+row)*FSTR + OFF + (c0+c)*M + m] += acc[r];
        }
      }
    __syncthreads();
  }
}

// ---------------------------------------------------------------------------
extern "C" void kernel_launch(void* const* d_in, const int* in_sizes, int n_in,
                              void* d_out, int out_size, void* d_ws, size_t ws_size,
                              hipStream_t stream) {
  if(n_in < 78) return;
  const float* POS; const int* Z; const float* EMB; const float* OW; const float* OB;
  BlockW BW[3];
  bool layA = (in_sizes[0] == NODES);   // insertion order: z first
  if(layA){
    Z   = (const int*)d_in[0];
    POS = (const float*)d_in[1];
    EMB = (const float*)d_in[3];
    for(int t=0;t<3;t++){
      int base = 4 + t*24;
      for(int l=0;l<4;l++){
        BW[t].WrA[l] = (const float*)d_in[base+l];
        BW[t].Ws[l]  = (const float*)d_in[base+4+l];
        BW[t].WrB[l] = (const float*)d_in[base+8+l];
        BW[t].V[l]   = (const float*)d_in[base+12+l];
        BW[t].LW[l]  = (const float*)d_in[base+16+l];
        BW[t].LB[l]  = (const float*)d_in[base+20+l];
      }
    }
    OW = (const float*)d_in[76]; OB = (const float*)d_in[77];
  } else {                              // jax pytree sorted-key order
    for(int t=0;t<3;t++){
      int base = 1 + t*24;
      for(int l=0;l<4;l++){
        BW[t].V[l]   = (const float*)d_in[base+l];
        BW[t].WrA[l] = (const float*)d_in[base+4+l];
        BW[t].WrB[l] = (const float*)d_in[base+8+l];
        BW[t].Ws[l]  = (const float*)d_in[base+12+l];
        BW[t].LB[l]  = (const float*)d_in[base+16+l];
        BW[t].LW[l]  = (const float*)d_in[base+20+l];
      }
    }
    EMB = (const float*)d_in[73]; OB = (const float*)d_in[74]; OW = (const float*)d_in[75];
    POS = (const float*)d_in[76]; Z  = (const int*)d_in[77];
  }

  // workspace (floats): 20*NF + 4*NODES*128  ~= 28.3 MB
  float* W   = (float*)d_ws;
  float* S0  = W;                                 // feats snapshots S(t)=W+t*NF, t=0..3
  float* GA  = W + 16*NF;
  float* GB  = W + 17*NF;
  float* GC  = W + 18*NF;
  float* DFB = W + 19*NF;
  float* SAB = W + 20*NF;                         // sA(t) = SAB + t*NODES*128
  float* DSA = SAB + (size_t)3*NODES*128;
  size_t totalF = 20*NF + (size_t)4*NODES*128;
  (void)ws_size; (void)out_size;

  float* Eout = (float*)d_out;                    // [0,1]=E ; +2 = F (B,N,3)
  float* Fout = Eout + 2;

  k_zero<<<1024,256,0,stream>>>(W, totalF);
  k_zero<<<32,256,0,stream>>>(Eout, 3074);
  k_embed<<<(NODES+127)/128,128,0,stream>>>(Z, EMB, S0);

  for(int t=0;t<3;t++){
    float* St  = W + (size_t)t*NF;
    float* St1 = W + (size_t)(t+1)*NF;
    float* CV  = W + (size_t)(4+t)*NF;
    float* HP  = W + (size_t)(7+t)*NF;
    float* ZP  = W + (size_t)(10+t)*NF;
    float* FBv = W + (size_t)(13+t)*NF;
    float* SAv = SAB + (size_t)t*NODES*128;
    k_sA_fB<<<NODES/16,32,0,stream>>>(St, BW[t], SAv, FBv);
    k_conv<<<NODES,128,0,stream>>>(POS, SAv, FBv, BW[t], CV);
    k_lin<<<NODES/16,32,0,stream>>>(CV, St, BW[t], HP, ZP, St1);
  }
  k_energy<<<(NODES+255)/256,256,0,stream>>>(W + 3*NF, OW, OB, Eout);

  k_init_g<<<(NODES+127)/128,128,0,stream>>>(OW, GA);
  float* gcur = GA; float* gnext = GB;
  for(int t=2;t>=0;t--){
    float* HP  = W + (size_t)(7+t)*NF;
    float* ZP  = W + (size_t)(10+t)*NF;
    float* FBv = W + (size_t)(13+t)*NF;
    float* SAv = SAB + (size_t)t*NODES*128;
    k_lin_bwd<<<NODES/16,32,0,stream>>>(gcur, ZP, HP, BW[t], GC, gnext);
    k_conv_bwdA<<<NODES,128,0,stream>>>(POS, SAv, FBv, GC, BW[t], Fout);
    k_conv_bwdB<<<NBATCH*(NN/16),128,0,stream>>>(POS, GC, BW[t], DSA, DFB);
    k_sAfB_bwd<<<NODES/16,32,0,stream>>>(DSA, DFB, BW[t], gnext);
    float* tmp = gcur; gcur = gnext; gnext = tmp;
  }
}